// GCN_88519275971050
// MI455X (gfx1250) — compile-verified
//
#include <hip/hip_runtime.h>
#include <hip/hip_bf16.h>

#define N_NODES     100000
#define N_EDGES     1000000
#define IN_FEATS    64
#define HID         128
#define NUM_CLASSES 10

typedef __attribute__((ext_vector_type(2))) float v2f;
typedef __attribute__((ext_vector_type(8))) float v8f;

// ---------------- degree / normalization ----------------

__global__ void deg_kernel(const long long* __restrict__ dst, float* __restrict__ deg) {
    int e = blockIdx.x * blockDim.x + threadIdx.x;
    if (e < N_EDGES) {
        __hip_atomic_fetch_add(&deg[(int)dst[e]], 1.0f,
                               __ATOMIC_RELAXED, __HIP_MEMORY_SCOPE_AGENT);
    }
}

__global__ void dinv_kernel(float* __restrict__ deg) {
    int i = blockIdx.x * blockDim.x + threadIdx.x;
    if (i < N_NODES) deg[i] = rsqrtf(deg[i] + 1.0f);   // +1 = self loop
}

// ---------------- fp32 WMMA GEMM: C[M,N] = A[M,K] @ B[K,N] ----------------
// One wave computes one 16x16 tile of C via V_WMMA_F32_16X16X4_F32.
// The block's 8 waves share one 16-column B tile, staged transposed in LDS
// with stride K+4 floats: lane n's b64 read hits banks 4n..4n+3 -> all 64
// banks covered once per wave, conflict-free, 8-byte aligned.
// With K,N compile-time constants and per-lane base pointers, the unrolled
// inner loop is: global_load_b64(A) + ds_load_b64(B) + v_wmma, immediate
// offsets only (no per-iteration address VALU).

template <int K, int N>
__global__ __launch_bounds__(256) void gemm_wmma(const float* __restrict__ A,
                                                 const float* __restrict__ B,
                                                 float* __restrict__ C, int M) {
    constexpr int KP = K + 4;
    __shared__ float sB[16 * KP];            // sB[n*KP + k] = B[k][col0+n]
    const int lane = threadIdx.x & 31;
    const int wave = threadIdx.x >> 5;
    const int col0 = blockIdx.y * 16;

    // cooperative transposed stage of the B tile
    for (int i = threadIdx.x; i < 16 * K; i += 256) {
        int k = i >> 4;
        int n = i & 15;
        sB[n * KP + k] = B[(size_t)k * N + (col0 + n)];
    }
    __syncthreads();

    const int rowTile = blockIdx.x * 8 + wave;
    if (rowTile * 16 >= M) return;           // wave-uniform: EXEC stays all-ones
    const int row0 = rowTile * 16;
    const int m  = lane & 15;                // A row / B,C column within tile
    const int hi = lane >> 4;                // half-wave -> K sub-pair

    const float* __restrict__ aptr = A + (size_t)(row0 + m) * K + 2 * hi;
    const float* sbp = &sB[m * KP + 2 * hi];

    v8f acc = {};
#pragma unroll
    for (int k0 = 0; k0 < K; k0 += 4) {
        v2f a = *(const v2f*)(aptr + k0);    // A[m][k0+2hi .. +1]
        v2f b = *(const v2f*)(sbp + k0);     // B[k0+2hi .. +1][n]
        acc = __builtin_amdgcn_wmma_f32_16x16x4_f32(
            /*neg_a=*/false, a, /*neg_b=*/false, b,
            /*c_mod=*/(short)0, acc, /*reuse_a=*/false, /*reuse_b=*/false);
    }

    // C tile: VGPR i -> row row0 + hi*8 + i, col col0 + m
    float* __restrict__ cptr = C + (size_t)(row0 + hi * 8) * N + (col0 + m);
#pragma unroll
    for (int i = 0; i < 8; ++i) cptr[i * N] = acc[i];
}

// ---------------- edge aggregation: out[dst] += h[src] * norm ----------------
// One wave per edge (32 lanes x float4 = 128 feats). Index/dinv loads are
// wave-uniform -> coalesce to broadcast. Atomics resolve in L2 (51MB
// accumulator + 51MB features fit the 192MB L2).

__global__ __launch_bounds__(256) void aggregate_kernel(const float* __restrict__ h,
                                                        const long long* __restrict__ src,
                                                        const long long* __restrict__ dst,
                                                        const float* __restrict__ dinv,
                                                        float* __restrict__ out) {
    long long t = (long long)blockIdx.x * blockDim.x + threadIdx.x;
    int e = (int)(t >> 5);
    if (e >= N_EDGES) return;
    int f = ((int)t & 31) * 4;
    int s = (int)src[e];
    int d = (int)dst[e];
    float norm = dinv[s] * dinv[d];
    const float4 hv = *(const float4*)(h + (size_t)s * HID + f);
    float* o = out + (size_t)d * HID + f;
    __hip_atomic_fetch_add(o + 0, hv.x * norm, __ATOMIC_RELAXED, __HIP_MEMORY_SCOPE_AGENT);
    __hip_atomic_fetch_add(o + 1, hv.y * norm, __ATOMIC_RELAXED, __HIP_MEMORY_SCOPE_AGENT);
    __hip_atomic_fetch_add(o + 2, hv.z * norm, __ATOMIC_RELAXED, __HIP_MEMORY_SCOPE_AGENT);
    __hip_atomic_fetch_add(o + 3, hv.w * norm, __ATOMIC_RELAXED, __HIP_MEMORY_SCOPE_AGENT);
}

// ---------------- self-loop + bias + ReLU (in place on agg) ----------------

__global__ void bias_relu_self(float* __restrict__ agg, const float* __restrict__ h,
                               const float* __restrict__ dinv,
                               const float* __restrict__ bias) {
    int t = blockIdx.x * blockDim.x + threadIdx.x;
    if (t >= N_NODES * HID) return;
    int node = t >> 7;          // /HID
    int j    = t & (HID - 1);
    float dv = dinv[node];
    float v  = agg[t] + h[t] * dv * dv + bias[j];
    agg[t] = v > 0.0f ? v : 0.0f;
}

// ---------------- classifier head + log_softmax ----------------

__global__ __launch_bounds__(256) void classifier_kernel(const float* __restrict__ h,
                                                         const float* __restrict__ Wl,
                                                         const float* __restrict__ bl,
                                                         float* __restrict__ out) {
    __shared__ float sW[HID * NUM_CLASSES];
    __shared__ float sB[NUM_CLASSES];
    for (int i = threadIdx.x; i < HID * NUM_CLASSES; i += blockDim.x) sW[i] = Wl[i];
    if (threadIdx.x < NUM_CLASSES) sB[threadIdx.x] = bl[threadIdx.x];
    __syncthreads();

    int node = blockIdx.x * blockDim.x + threadIdx.x;
    if (node >= N_NODES) return;

    float z[NUM_CLASSES];
#pragma unroll
    for (int c = 0; c < NUM_CLASSES; ++c) z[c] = sB[c];

    const float* __restrict__ hr = h + (size_t)node * HID;
    for (int k = 0; k < HID; ++k) {
        float xv = hr[k];
#pragma unroll
        for (int c = 0; c < NUM_CLASSES; ++c) z[c] += xv * sW[k * NUM_CLASSES + c];
    }

    float mx = z[0];
#pragma unroll
    for (int c = 1; c < NUM_CLASSES; ++c) mx = fmaxf(mx, z[c]);
    float sum = 0.0f;
#pragma unroll
    for (int c = 0; c < NUM_CLASSES; ++c) sum += __expf(z[c] - mx);
    float lse = mx + __logf(sum);

    float* o = out + (size_t)node * NUM_CLASSES;
#pragma unroll
    for (int c = 0; c < NUM_CLASSES; ++c) o[c] = z[c] - lse;
}

// ---------------- driver ----------------
// d_in order: x, edge_index(int64, [2,E]), W1, b1, W2, b2, Wl, bl
// Workspace layout: bufA[51.2MB] | bufB[51.2MB] | dinv[400KB]  (~103MB total)

extern "C" void kernel_launch(void* const* d_in, const int* in_sizes, int n_in,
                              void* d_out, int out_size, void* d_ws, size_t ws_size,
                              hipStream_t stream) {
    const float*     x   = (const float*)d_in[0];
    const long long* ei  = (const long long*)d_in[1];
    const long long* src = ei;                 // edge_index[0]
    const long long* dst = ei + N_EDGES;       // edge_index[1]
    const float* W1 = (const float*)d_in[2];
    const float* b1 = (const float*)d_in[3];
    const float* W2 = (const float*)d_in[4];
    const float* b2 = (const float*)d_in[5];
    const float* Wl = (const float*)d_in[6];
    const float* bl = (const float*)d_in[7];
    float* out = (float*)d_out;

    const size_t szh = (size_t)N_NODES * HID * sizeof(float);
    float* bufA = (float*)d_ws;
    float* bufB = (float*)((char*)d_ws + szh);
    float* dinv = (float*)((char*)d_ws + 2 * szh);

    // --- normalization coefficients ---
    hipMemsetAsync(dinv, 0, N_NODES * sizeof(float), stream);
    deg_kernel<<<(N_EDGES + 255) / 256, 256, 0, stream>>>(dst, dinv);
    dinv_kernel<<<(N_NODES + 255) / 256, 256, 0, stream>>>(dinv);

    const dim3 gemmGrid((N_NODES / 16 + 7) / 8, HID / 16);   // 782 x 8 blocks, 8 waves each
    const unsigned aggBlocks = (unsigned)(((long long)N_EDGES * 32 + 255) / 256);
    const unsigned ewBlocks  = (N_NODES * HID + 255) / 256;

    // --- layer 1: h1 = x @ W1 ; agg ; relu(agg + self + b1) ---
    gemm_wmma<IN_FEATS, HID><<<gemmGrid, 256, 0, stream>>>(x, W1, bufA, N_NODES);
    hipMemsetAsync(bufB, 0, szh, stream);
    aggregate_kernel<<<aggBlocks, 256, 0, stream>>>(bufA, src, dst, dinv, bufB);
    bias_relu_self<<<ewBlocks, 256, 0, stream>>>(bufB, bufA, dinv, b1);

    // --- layer 2: h2 = out1 @ W2 ; agg ; relu(agg + self + b2) ---
    gemm_wmma<HID, HID><<<gemmGrid, 256, 0, stream>>>(bufB, W2, bufA, N_NODES);
    hipMemsetAsync(bufB, 0, szh, stream);
    aggregate_kernel<<<aggBlocks, 256, 0, stream>>>(bufA, src, dst, dinv, bufB);
    bias_relu_self<<<ewBlocks, 256, 0, stream>>>(bufB, bufA, dinv, b2);

    // --- head: log_softmax(out2 @ Wl + bl) ---
    classifier_kernel<<<(N_NODES + 255) / 256, 256, 0, stream>>>(bufB, Wl, bl, out);
}